// DeepGCNLayer_55027120996500
// MI455X (gfx1250) — compile-verified
//
#include <hip/hip_runtime.h>
#include <hip/hip_bf16.h>

#define EMB 128
#define BN_EPS 1e-5f

typedef __attribute__((ext_vector_type(2))) float v2f;
typedef __attribute__((ext_vector_type(8))) float v8f;

// ---------------------------------------------------------------------------
// K0: zero workspace region (agg + degrees + BN stats), float4 stores.
// ---------------------------------------------------------------------------
__global__ __launch_bounds__(256) void k_zero(float4* ws, size_t n4) {
    size_t i = (size_t)blockIdx.x * 256 + threadIdx.x;
    if (i < n4) ws[i] = make_float4(0.f, 0.f, 0.f, 0.f);
}

// ---------------------------------------------------------------------------
// K1: degree accumulation (float atomics; counts are small integers -> exact)
// ---------------------------------------------------------------------------
__global__ __launch_bounds__(256) void k_degree(const int* __restrict__ src,
                                                const int* __restrict__ dst,
                                                float* __restrict__ deg_out,
                                                float* __restrict__ deg_in,
                                                int n_edges) {
    int e = blockIdx.x * 256 + threadIdx.x;
    if (e >= n_edges) return;
    atomicAdd(&deg_out[src[e]], 1.0f);
    atomicAdd(&deg_in[dst[e]], 1.0f);
}

// ---------------------------------------------------------------------------
// K2: deg -> rsqrt(max(deg,1)) in place (becomes norm_src / norm_dst)
// ---------------------------------------------------------------------------
__global__ __launch_bounds__(256) void k_norm(float* __restrict__ deg_out,
                                              float* __restrict__ deg_in,
                                              int n_nodes) {
    int i = blockIdx.x * 256 + threadIdx.x;
    if (i >= n_nodes) return;
    deg_out[i] = __frsqrt_rn(fmaxf(deg_out[i], 1.0f));
    deg_in[i]  = __frsqrt_rn(fmaxf(deg_in[i], 1.0f));
}

// ---------------------------------------------------------------------------
// K3: edge scatter in x-space: agg[dst] += x[src] * norm_src[src]
// One wave (32 lanes) per edge; float4 per lane covers all 128 features.
// Bandwidth-dominant kernel (~1.6 GB of traffic -> ~70-100 us at 23.3 TB/s).
// ---------------------------------------------------------------------------
__global__ __launch_bounds__(256) void k_scatter(const float* __restrict__ x,
                                                 const int* __restrict__ src,
                                                 const int* __restrict__ dst,
                                                 const float* __restrict__ norm_src,
                                                 float* __restrict__ agg,
                                                 int n_edges) {
    int wave = threadIdx.x >> 5;
    int lane = threadIdx.x & 31;
    int e = blockIdx.x * 8 + wave;
    if (e >= n_edges) return;
    int s = src[e];                       // wave-uniform -> scalar loads
    int d = dst[e];
    float ns = norm_src[s];
    const float4* xp = (const float4*)(x + (size_t)s * EMB);
    float4 v = xp[lane];
    float* ap = agg + (size_t)d * EMB + lane * 4;
    atomicAdd(ap + 0, v.x * ns);
    atomicAdd(ap + 1, v.y * ns);
    atomicAdd(ap + 2, v.z * ns);
    atomicAdd(ap + 3, v.w * ns);
}

// ---------------------------------------------------------------------------
// K4: h2 = (agg @ W) * norm_dst + b  via V_WMMA_F32_16X16X4_F32.
// 256 threads = 8 waves; each wave computes a 16x128 output tile.
//
// W is staged in LDS in K-PAIR-INTERLEAVED layout:
//   Wlds2[kp * 128 + n] = float2{ W[2*kp][n], W[2*kp+1][n] }   (64 KB)
// so each WMMA B fragment is a single 8-byte ds_load_b64 into an aligned
// VGPR pair -- no repacking movs.
//
// A-frag (16x4, ISA 7.12.2): lanes 0-15 -> M=lane,    K={k0,k0+1};
//                            lanes 16-31 -> M=lane-16, K={k0+2,k0+3}.
// B-frag (4x16): lanes stripe N; lane-half h reads k-pair (k0/2 + h).
// C/D: VGPR r holds (M = r + 8*half, N = lane&15).
//
// Boundary handling: A loads are UNCONDITIONAL from a clamped row pointer,
// multiplied by a 0/1 mask (one v_pk_mul_f32) -> EXEC stays all-ones around
// WMMA as the ISA requires.
// ---------------------------------------------------------------------------
__global__ __launch_bounds__(256) void k_gemm_wmma(const float* __restrict__ agg,
                                                   const float* __restrict__ W,
                                                   const float* __restrict__ bias,
                                                   const float* __restrict__ norm_dst,
                                                   float* __restrict__ out,
                                                   int n_nodes) {
    __shared__ v2f Wlds2[(EMB / 2) * EMB]; // 64 KB, k-pair interleaved

    // cooperative interleaving load: 8 iters x 256 threads x 4 float2
    {
        const float4* Wg4 = (const float4*)W;
        #pragma unroll
        for (int i = 0; i < 8; ++i) {
            const int g   = i * 256 + threadIdx.x;   // float2-group id, 0..2047
            const int kp  = g >> 5;                  // k-pair 0..63
            const int n4  = (g & 31) * 4;            // n base 0..124
            const float4 r0 = Wg4[(2 * kp) * (EMB / 4) + (n4 >> 2)];
            const float4 r1 = Wg4[(2 * kp + 1) * (EMB / 4) + (n4 >> 2)];
            v2f* dp = &Wlds2[kp * EMB + n4];
            dp[0] = (v2f){r0.x, r1.x};
            dp[1] = (v2f){r0.y, r1.y};
            dp[2] = (v2f){r0.z, r1.z};
            dp[3] = (v2f){r0.w, r1.w};
        }
    }
    __syncthreads();

    const int wave = threadIdx.x >> 5;
    const int lane = threadIdx.x & 31;
    const int half = lane >> 4;     // 0: lanes 0-15, 1: lanes 16-31
    const int l16  = lane & 15;
    const int m0   = (blockIdx.x * 8 + wave) * 16;

    v8f acc[8];
    #pragma unroll
    for (int nt = 0; nt < 8; ++nt) acc[nt] = (v8f){0,0,0,0,0,0,0,0};

    const int   arow  = m0 + l16;
    const float amask = (arow < n_nodes) ? 1.0f : 0.0f;
    const int   srow  = (arow < n_nodes) ? arow : (n_nodes - 1);
    const float* aptr = agg + (size_t)srow * EMB + 2 * half;

    __builtin_prefetch(aptr, 0, 0);           // global_prefetch_b8 on the A row

    // per-lane-half B base: k-pair (k0/2 + half), column l16
    const v2f* bbase = &Wlds2[half * EMB + l16];

    #pragma unroll 2
    for (int k0 = 0; k0 < EMB; k0 += 4) {
        v2f a = *(const v2f*)(aptr + k0);     // single global_load_b64
        a.x *= amask;
        a.y *= amask;
        const v2f* bk = bbase + (k0 >> 1) * EMB;
        #pragma unroll
        for (int nt = 0; nt < 8; ++nt) {
            const v2f bf = bk[nt * 16];       // single ds_load_b64
            acc[nt] = __builtin_amdgcn_wmma_f32_16x16x4_f32(
                false, a, false, bf, (short)0, acc[nt], false, false);
        }
    }

    // store with fused norm_dst + bias
    #pragma unroll
    for (int r = 0; r < 8; ++r) {
        const int m = m0 + r + 8 * half;
        if (m < n_nodes) {
            const float nd = norm_dst[m];
            float* orow = out + (size_t)m * EMB;
            #pragma unroll
            for (int nt = 0; nt < 8; ++nt) {
                const int n = nt * 16 + l16;
                orow[n] = acc[nt][r] * nd + bias[n];
            }
        }
    }
}

// ---------------------------------------------------------------------------
// K5: BN column statistics: sums[c] = sum_m h2[m][c], sumsq[c] = sum h2^2
// Block reads 2 contiguous rows (1KB) per iteration -> fully coalesced.
// ---------------------------------------------------------------------------
__global__ __launch_bounds__(256) void k_bn_stats(const float* __restrict__ h,
                                                  float* __restrict__ sums,
                                                  float* __restrict__ sumsq,
                                                  int n_nodes) {
    const int col  = threadIdx.x & (EMB - 1);
    const int rsub = threadIdx.x >> 7; // 0 or 1
    float s = 0.0f, q = 0.0f;
    for (int row = blockIdx.x * 2 + rsub; row < n_nodes; row += gridDim.x * 2) {
        float v = h[(size_t)row * EMB + col];
        s += v;
        q += v * v;
    }
    __shared__ float ls[256];
    __shared__ float lq[256];
    ls[threadIdx.x] = s;
    lq[threadIdx.x] = q;
    __syncthreads();
    if (rsub == 0) {
        atomicAdd(&sums[col],  ls[col] + ls[col + EMB]);
        atomicAdd(&sumsq[col], lq[col] + lq[col + EMB]);
    }
}

// ---------------------------------------------------------------------------
// K6: out = relu((h2 - mean) * rsqrt(var+eps) * gamma + beta) + x  (in place)
// float4 per thread: 3 x 16B vector memory ops instead of 12 scalar ones.
// ---------------------------------------------------------------------------
__global__ __launch_bounds__(256) void k_finalize(const float4* __restrict__ x4,
                                                  const float4* __restrict__ gamma4,
                                                  const float4* __restrict__ beta4,
                                                  const float4* __restrict__ sums4,
                                                  const float4* __restrict__ sumsq4,
                                                  float4* __restrict__ out4,
                                                  size_t total4, float inv_n) {
    size_t idx = (size_t)blockIdx.x * 256 + threadIdx.x;
    if (idx >= total4) return;
    const int cg = (int)(idx & (EMB / 4 - 1));   // float4 column group 0..31

    const float4 s  = sums4[cg];
    const float4 q  = sumsq4[cg];
    const float4 g  = gamma4[cg];
    const float4 bt = beta4[cg];
    const float4 h  = out4[idx];
    const float4 xr = x4[idx];

    float4 r;
    {
        float mean = s.x * inv_n, var = q.x * inv_n - mean * mean;
        r.x = fmaxf((h.x - mean) * __frsqrt_rn(var + BN_EPS) * g.x + bt.x, 0.f) + xr.x;
    }
    {
        float mean = s.y * inv_n, var = q.y * inv_n - mean * mean;
        r.y = fmaxf((h.y - mean) * __frsqrt_rn(var + BN_EPS) * g.y + bt.y, 0.f) + xr.y;
    }
    {
        float mean = s.z * inv_n, var = q.z * inv_n - mean * mean;
        r.z = fmaxf((h.z - mean) * __frsqrt_rn(var + BN_EPS) * g.z + bt.z, 0.f) + xr.z;
    }
    {
        float mean = s.w * inv_n, var = q.w * inv_n - mean * mean;
        r.w = fmaxf((h.w - mean) * __frsqrt_rn(var + BN_EPS) * g.w + bt.w, 0.f) + xr.w;
    }
    out4[idx] = r;
}

// ---------------------------------------------------------------------------
// Host launcher
// ---------------------------------------------------------------------------
extern "C" void kernel_launch(void* const* d_in, const int* in_sizes, int n_in,
                              void* d_out, int out_size, void* d_ws, size_t ws_size,
                              hipStream_t stream) {
    const float* x     = (const float*)d_in[0];
    const int*   src   = (const int*)d_in[1];
    const int*   dst   = (const int*)d_in[2];
    const float* W     = (const float*)d_in[3];
    const float* b     = (const float*)d_in[4];
    const float* gamma = (const float*)d_in[5];
    const float* beta  = (const float*)d_in[6];
    float* out = (float*)d_out;

    const int n_nodes = in_sizes[0] / EMB;
    const int n_edges = in_sizes[1];

    // workspace layout (floats): agg[N*128] | deg_out[N] | deg_in[N] | sums[128] | sumsq[128]
    float* ws      = (float*)d_ws;
    float* agg     = ws;
    float* deg_out = agg + (size_t)n_nodes * EMB;
    float* deg_in  = deg_out + n_nodes;
    float* sums    = deg_in + n_nodes;
    float* sumsq   = sums + EMB;

    const size_t zero_n  = (size_t)n_nodes * EMB + 2 * (size_t)n_nodes + 2 * EMB;
    const size_t zero_n4 = (zero_n + 3) / 4;   // all segments are multiples of 4 floats

    k_zero<<<(unsigned)((zero_n4 + 255) / 256), 256, 0, stream>>>((float4*)ws, zero_n4);

    k_degree<<<(n_edges + 255) / 256, 256, 0, stream>>>(src, dst, deg_out, deg_in, n_edges);

    k_norm<<<(n_nodes + 255) / 256, 256, 0, stream>>>(deg_out, deg_in, n_nodes);

    // scatter: 8 edges (waves) per block
    k_scatter<<<(n_edges + 7) / 8, 256, 0, stream>>>(x, src, dst, deg_out, agg, n_edges);

    // GEMM: 128 rows per block (8 waves x 16 rows)
    k_gemm_wmma<<<(n_nodes + 127) / 128, 256, 0, stream>>>(agg, W, b, deg_in, out, n_nodes);

    k_bn_stats<<<512, 256, 0, stream>>>(out, sums, sumsq, n_nodes);

    const size_t total4 = (size_t)n_nodes * EMB / 4;
    k_finalize<<<(unsigned)((total4 + 255) / 256), 256, 0, stream>>>(
        (const float4*)x, (const float4*)gamma, (const float4*)beta,
        (const float4*)sums, (const float4*)sumsq, (float4*)out,
        total4, 1.0f / (float)n_nodes);
}